// RNNPolicy_23450521436692
// MI455X (gfx1250) — compile-verified
//
#include <hip/hip_runtime.h>

// Sizes fixed by the reference.
#define HID   32
#define IN_CH 128
#define OUTD  8
#define BATCH 512
#define TLEN  512

typedef float v2f __attribute__((ext_vector_type(2)));
typedef float v4f __attribute__((ext_vector_type(4)));
typedef float v8f __attribute__((ext_vector_type(8)));

// ---------------------------------------------------------------------------
// Phase 1: xp[b,t,:] = seq[b,t,:] @ W_ih^T + (b_ih + b_hh)
// One wave per (16-batch tile, t). Memory-bound: 128 MB of seq @ 23.3 TB/s
// ~ 5.5 us floor, so exact f32 WMMA (16x16x4) and the job is feeding it.
// seq loads are NON-TEMPORAL so the one-shot 128 MB stream does not evict
// the 32 MB xp intermediate from the 192 MB L2 (phase 2 re-reads xp).
// D-tiles are stored in native WMMA C-register order so phase 2 reloads
// accumulators with coalesced b128 loads - correct by construction.
// ---------------------------------------------------------------------------
__global__ __launch_bounds__(256) void rnn_xproj_kernel(
    const float* __restrict__ seq,
    const float* __restrict__ W_ih,
    const float* __restrict__ b_ih,
    const float* __restrict__ b_hh,
    float* __restrict__ xp_ws)
{
    // W_ih staged in LDS, row stride 132 dwords -> bank = (4*row + col)%64 for
    // the B-operand access pattern: conflict-free b64 reads.
    __shared__ float lds_w[HID * 132];
    __shared__ float lds_bias[HID];

    const int tid = threadIdx.x;
    for (int idx = tid; idx < HID * IN_CH; idx += 256) {
        int row = idx >> 7, col = idx & 127;
        lds_w[row * 132 + col] = W_ih[idx];
    }
    if (tid < HID) lds_bias[tid] = b_ih[tid] + b_hh[tid];
    __syncthreads();

    const int lane = tid & 31;
    const int wave = tid >> 5;
    const int tile = blockIdx.x * 8 + wave;      // 0 .. 16383
    const int t    = tile & (TLEN - 1);
    const int bt   = tile >> 9;                  // batch tile 0..31
    const int m    = lane & 15;                  // A-matrix row (batch within tile)
    const int half = lane >> 4;                  // K sub-pair select (ISA A layout)

    // A operand: seq rows for this tile, 32 chunks of 16x4 f32.
    // Lane L holds A[m][4k + 2*half + {0,1}] per the 16x4 f32 A layout.
    // Non-temporal: streamed exactly once, keep L2 for xp.
    v2f a[32];
    const float* arow = seq + ((size_t)(bt * 16 + m) * TLEN + t) * IN_CH + 2 * half;
#pragma unroll
    for (int kc = 0; kc < 32; ++kc)
        a[kc] = __builtin_nontemporal_load((const v2f*)(arow + 4 * kc));

    float* outp = xp_ws + (size_t)tile * 512;    // 2 N-tiles * 256 floats

#pragma unroll
    for (int nt = 0; nt < 2; ++nt) {
        // Two independent partial chains per N-tile -> 4 concurrent WMMA
        // chains in flight across the nt loop; summed at the end.
        v8f accA = {}, accB = {};
        const float* wrow = lds_w + (nt * 16 + m) * 132 + 2 * half;
#pragma unroll
        for (int kc = 0; kc < 16; ++kc) {
            v2f bA = *(const v2f*)(wrow + 4 * kc);
            v2f bB = *(const v2f*)(wrow + 4 * (kc + 16));
            accA = __builtin_amdgcn_wmma_f32_16x16x4_f32(
                false, a[kc], false, bA, (short)0, accA, false, false);
            accB = __builtin_amdgcn_wmma_f32_16x16x4_f32(
                false, a[kc + 16], false, bB, (short)0, accB, false, false);
        }
        // Fused bias: lane L holds column n = L%16 of the C tile.
        float bias = lds_bias[nt * 16 + m];
        v8f acc;
#pragma unroll
        for (int i = 0; i < 8; ++i) acc[i] = accA[i] + accB[i] + bias;

        // Store D in register order: wave writes a contiguous 1 KB block.
        float* po = outp + nt * 256 + lane * 8;
        *(v4f*)(po)     = v4f{acc[0], acc[1], acc[2], acc[3]};
        *(v4f*)(po + 4) = v4f{acc[4], acc[5], acc[6], acc[7]};
    }
}

// ---------------------------------------------------------------------------
// Phase 2: h = relu(xp_t + h @ W_hh^T) scanned over t; then FC head.
// Latency-bound sequential scan. Per 16-batch tile use TWO waves (one per
// 16-column N-tile) sharing h through LDS, and within each wave split K into
// two independent 4-deep WMMA chains combined at ReLU time. xp(t+1) is
// software-prefetched so its L2 latency hides under step t's math.
// ---------------------------------------------------------------------------
__global__ __launch_bounds__(64) void rnn_scan_kernel(
    const float* __restrict__ xp_ws,
    const float* __restrict__ W_hh,
    const float* __restrict__ W_fc,
    const float* __restrict__ b_fc,
    float* __restrict__ out)
{
    // h buffer: [m 0..15][hidden 0..31], row stride 34 dwords:
    // keeps b64 alignment (even offsets) and near conflict-free banks.
    __shared__ float lds_h[16 * 34];

    const int tid  = threadIdx.x;
    const int lane = tid & 31;
    const int nt   = tid >> 5;                   // wave id = N-tile (0 or 1)
    const int bt   = blockIdx.x;                 // batch tile 0..31
    const int m    = lane & 15;
    const int half = lane >> 4;

    // Preload this wave's W_hh B-operand chunks: B[k][n] = W_hh[nt*16 + n][k].
    v2f bw[8];
#pragma unroll
    for (int kc = 0; kc < 8; ++kc)
        bw[kc] = *(const v2f*)(W_hh + (nt * 16 + m) * HID + 4 * kc + 2 * half);

    // h A-operand chunks, h0 = 0.
    v2f ha[8];
#pragma unroll
    for (int kc = 0; kc < 8; ++kc) ha[kc] = v2f{0.f, 0.f};

    const float* cbase = xp_ws + (size_t)bt * TLEN * 512 + nt * 256 + lane * 8;

    // Preload xp(t=0) accumulator tile (native C-register order from phase 1).
    v4f clo = *(const v4f*)(cbase);
    v4f chi = *(const v4f*)(cbase + 4);

    for (int t = 0; t < TLEN; ++t) {
        // Prefetch xp(t+1) early; chain below hides the L2 latency.
        int tn = (t + 1 < TLEN) ? (t + 1) : t;
        v4f nlo = *(const v4f*)(cbase + (size_t)tn * 512);
        v4f nhi = *(const v4f*)(cbase + (size_t)tn * 512 + 4);

        // Two independent 4-deep WMMA chains over K=32.
        v8f p0 = {}, p1 = {};
#pragma unroll
        for (int kc = 0; kc < 4; ++kc) {
            p0 = __builtin_amdgcn_wmma_f32_16x16x4_f32(
                false, ha[kc], false, bw[kc], (short)0, p0, false, false);
            p1 = __builtin_amdgcn_wmma_f32_16x16x4_f32(
                false, ha[kc + 4], false, bw[kc + 4], (short)0, p1, false, false);
        }
        // h = relu(xp + p0 + p1)
        v8f acc;
#pragma unroll
        for (int i = 0; i < 4; ++i) {
            acc[i]     = fmaxf(clo[i] + p0[i]     + p1[i],     0.f);
            acc[i + 4] = fmaxf(chi[i] + p0[i + 4] + p1[i + 4], 0.f);
        }

        // C-layout -> LDS [m][hidden]: lane L holds D[r + 8*half][L%16].
        __syncthreads();   // previous step's A-loads done before overwrite
#pragma unroll
        for (int r = 0; r < 8; ++r)
            lds_h[(r + 8 * half) * 34 + nt * 16 + m] = acc[r];
        __syncthreads();

        // LDS -> A-layout for next step: lane L reads row m, cols 4k+2*half.
#pragma unroll
        for (int kc = 0; kc < 8; ++kc)
            ha[kc] = *(const v2f*)(lds_h + m * 34 + 4 * kc + 2 * half);

        clo = nlo; chi = nhi;
    }
    __syncthreads();

    // FC head: out[b0+m][o] = h_last[m,:] . W_fc[o,:] + b_fc[o]  (tiny; VALU)
    for (int idx = tid; idx < 16 * OUTD; idx += 64) {
        int mm = idx >> 3, o = idx & 7;
        float s = b_fc[o];
#pragma unroll
        for (int h = 0; h < HID; ++h)
            s += lds_h[mm * 34 + h] * W_fc[o * HID + h];
        out[(bt * 16 + mm) * OUTD + o] = s;
    }
}

// ---------------------------------------------------------------------------
extern "C" void kernel_launch(void* const* d_in, const int* in_sizes, int n_in,
                              void* d_out, int out_size, void* d_ws, size_t ws_size,
                              hipStream_t stream) {
    const float* seq  = (const float*)d_in[0];
    const float* W_ih = (const float*)d_in[1];
    const float* W_hh = (const float*)d_in[2];
    const float* b_ih = (const float*)d_in[3];
    const float* b_hh = (const float*)d_in[4];
    const float* W_fc = (const float*)d_in[5];
    const float* b_fc = (const float*)d_in[6];
    float* outp = (float*)d_out;
    float* xp   = (float*)d_ws;   // 512*512*32 floats = 32 MB (fits in L2)

    // Phase 1: 16384 tiles, 8 waves/block -> 2048 blocks of 256 threads.
    rnn_xproj_kernel<<<2048, 256, 0, stream>>>(seq, W_ih, b_ih, b_hh, xp);
    // Phase 2: 32 batch tiles, two waves (one per N-tile) each.
    rnn_scan_kernel<<<32, 64, 0, stream>>>(xp, W_hh, W_fc, b_fc, outp);
}